// LinearStateSpaceLayer_88837103550641
// MI455X (gfx1250) — compile-verified
//
#include <hip/hip_runtime.h>

// ---------------------------------------------------------------------------
// S4/HiPPO linear state-space layer for MI455X (gfx1250).
//   h_{t+1} = Ab h_t + Bb x_t ; y_t = C h_{t+1},  N=512, L=16384
// Strategy: triangular solve for Ab/Bb (A is lower triangular), chunked
// parallel scan with fp32 WMMA (v_wmma_f32_16x16x4_f32) GEMMs, Hillis-Steele
// combine using repeated-squared powers of Ab^CT. fp32 WMMA keeps the
// 16384-step recurrence numerically faithful to the fp32 reference; all
// matrix traffic is L2-resident (Ab = 1MB << 192MB L2), so this is matrix-
// pipe bound, not HBM bound (23.3 TB/s is irrelevant here).
// Prefetches (global_prefetch_b8) are straight-line, BEFORE the K-loops:
// in-loop prefetch conditionals were shown (rounds 2-3) to break the
// compiler's 21-deep load pipelining of the WMMA stream.
// ---------------------------------------------------------------------------

#define Ndim 512
#define CT   16          // chunk length (NC = L/CT chunks)

typedef float v2f __attribute__((ext_vector_type(2)));
typedef float v8f __attribute__((ext_vector_type(8)));

__device__ __forceinline__ v8f wmma_f32_16x16x4(v2f a, v2f b, v8f c) {
  // 8-arg form: (neg_a, A, neg_b, B, c_mod, C, reuse_a, reuse_b)
  return __builtin_amdgcn_wmma_f32_16x16x4_f32(false, a, false, b,
                                               (short)0, c, false, false);
}

__device__ __forceinline__ v8f v8f_zero() {
  v8f z = {0.f, 0.f, 0.f, 0.f, 0.f, 0.f, 0.f, 0.f};
  return z;
}

// global_prefetch_b8 (cold-start warming only; never inside hot loops)
__device__ __forceinline__ void prefetch_ro(const void* p) {
  __builtin_prefetch(p, 0, 1);
}

// ---------------------------------------------------------------------------
// Kernel 1: forward substitution. A is lower triangular, so
//   M = I - (dt/2) A  is lower triangular.  Solve M * Ab = I + (dt/2) A
// column by column (one wave per column), plus one extra column for
//   M * Bb = dt * B.
// ---------------------------------------------------------------------------
__global__ __launch_bounds__(256) void build_ab_bb(
    const float* __restrict__ A, const float* __restrict__ Bv,
    float* __restrict__ Ab, float* __restrict__ Bb, float dt)
{
  __shared__ float ylds[8][Ndim];
  const int wave = threadIdx.x >> 5;
  const int lane = threadIdx.x & 31;
  const int j = blockIdx.x * 8 + wave;      // column id, 0..512
  if (j > Ndim) return;                      // wave-uniform exit
  float* yl = ylds[wave];
  const float hd = 0.5f * dt;
  const bool isB = (j == Ndim);
  const int start = isB ? 0 : j;             // Ab column j is zero above row j

  for (int k = lane; k < Ndim; k += 32) yl[k] = 0.0f;
  __builtin_amdgcn_wave_barrier();

  for (int i = start; i < Ndim; ++i) {
    const float* Arow = A + i * Ndim;
    float partial = 0.0f;                    // sum_k A[i,k]*y[k], k in [start,i)
    for (int k = start + lane; k < i; k += 32) partial += Arow[k] * yl[k];
    #pragma unroll
    for (int off = 16; off > 0; off >>= 1) partial += __shfl_xor(partial, off, 32);
    const float rhs = isB ? (dt * Bv[i])
                          : ((i == j ? 1.0f : 0.0f) + hd * Arow[j]);
    const float Mii = 1.0f - hd * Arow[i];
    // M[i,k<i] = -hd*A[i,k]  =>  y_i = (rhs + hd*partial) / Mii
    const float yi = (rhs + hd * partial) / Mii;
    if (lane == 0) yl[i] = yi;
    __builtin_amdgcn_wave_barrier();
  }
  __builtin_amdgcn_wave_barrier();

  if (isB) {
    for (int i = lane; i < Ndim; i += 32) Bb[i] = yl[i];
  } else {
    for (int i = lane; i < Ndim; i += 32)
      Ab[i * Ndim + j] = (i < start) ? 0.0f : yl[i];
  }
}

// ---------------------------------------------------------------------------
// Kernel 2: D = S @ S  (512x512x512 fp32 WMMA GEMM) for repeated squaring.
// One 16x16 output tile per wave; grid = (32 M-tiles, 4) with 8 waves/WG.
// Flat K-loop (no pragma): compiler unrolls by 8 and keeps ~21 loads in
// flight across the WMMA stream (verified round 1).
// ---------------------------------------------------------------------------
__global__ __launch_bounds__(256) void mat_square(
    const float* __restrict__ S, float* __restrict__ D)
{
  const int wave = threadIdx.x >> 5;
  const int lane = threadIdx.x & 31;
  const int ln15 = lane & 15;                // A-row / B-col / D-col
  const int kidx = (lane >> 4) * 2;          // K-pair selector (lanes 16-31 -> K+2)
  const int moff = (lane >> 4) * 8;          // D rows +8 for upper half-wave
  const int M0 = blockIdx.x * 16;
  const int N0 = (blockIdx.y * 8 + wave) * 16;

  // straight-line cold-start prefetch of the A row block (one line per lane)
  prefetch_ro(&S[(M0 + ln15) * Ndim]);
  prefetch_ro(&S[(M0 + ln15) * Ndim + 128]);
  prefetch_ro(&S[(M0 + ln15) * Ndim + 256]);
  prefetch_ro(&S[(M0 + ln15) * Ndim + 384]);

  v8f acc = v8f_zero();
  for (int kk = 0; kk < Ndim; kk += 4) {
    v2f a, b;
    a.x = S[(M0 + ln15) * Ndim + kk + kidx];
    a.y = S[(M0 + ln15) * Ndim + kk + kidx + 1];
    b.x = S[(kk + kidx)     * Ndim + N0 + ln15];
    b.y = S[(kk + kidx + 1) * Ndim + N0 + ln15];
    acc = wmma_f32_16x16x4(a, b, acc);
  }
  #pragma unroll
  for (int r = 0; r < 8; ++r)
    D[(M0 + moff + r) * Ndim + N0 + ln15] = acc[r];
}

// ---------------------------------------------------------------------------
// Kernel 3: chunk-local scan, batched across 16 chunks per WG (columns of a
// 512x16 state matrix S in LDS). Each local step is S <- Ab@S + Bb (x) x_row,
// done as WMMA GEMM: 8 waves x 4 M-tiles each cover all 512 rows, N=16 cols.
// pass1: Hprev=null, Vout=end states.  pass2: Hprev=scan result (init states),
// Cvec/yout produce outputs; y column-dots reduced via shfl + LDS ds_add.
// ---------------------------------------------------------------------------
__global__ __launch_bounds__(256) void chunk_pass(
    const float* __restrict__ Ab, const float* __restrict__ Bb,
    const float* __restrict__ x,  const float* __restrict__ h0,
    const float* __restrict__ Hprev,   // init states (pass2) or null
    float* __restrict__ Vout,          // end states out (pass1) or null
    const float* __restrict__ Cvec,    // C row (pass2) or null
    float* __restrict__ yout,          // y out (pass2) or null
    int NC)
{
  __shared__ float Sb[Ndim * 16];    // state: [m][col], 32 KB
  __shared__ float Bbl[Ndim];
  __shared__ float Cl[Ndim];
  __shared__ float yacc[16];

  const int tid  = threadIdx.x;
  const int wave = tid >> 5, lane = tid & 31;
  const int col  = lane & 15;
  const int kidx = (lane >> 4) * 2;
  const int moff = (lane >> 4) * 8;
  const int cgBase = blockIdx.x * 16;

  // cold-start prefetch: warm this wave's four Ab row-tiles (t=0 misses)
  #pragma unroll
  for (int tt = 0; tt < 4; ++tt) {
    const float* r0 = &Ab[(wave * 64 + tt * 16 + col) * Ndim];
    prefetch_ro(r0);
    prefetch_ro(r0 + 256);
  }

  for (int m = tid; m < Ndim; m += 256) {
    Bbl[m] = Bb[m];
    Cl[m]  = Cvec ? Cvec[m] : 0.0f;
  }
  if (tid < 16) yacc[tid] = 0.0f;
  // initial state per column: chunk 0 starts from h0, others from scan result
  for (int e = tid; e < Ndim * 16; e += 256) {
    const int m = e >> 4, c = e & 15;
    const int cg = cgBase + c;
    float v = 0.0f;
    if (cg == 0)        { if (h0) v = h0[m]; }
    else if (Hprev)     { v = Hprev[m * NC + (cg - 1)]; }
    Sb[m * 16 + c] = v;
  }
  __syncthreads();

  for (int t = 0; t < CT; ++t) {
    const float xv = x[(cgBase + col) * CT + t];
    v8f acc[4];
    #pragma unroll
    for (int tt = 0; tt < 4; ++tt) acc[tt] = v8f_zero();

    for (int kk = 0; kk < Ndim; kk += 4) {
      v2f b;
      b.x = Sb[(kk + kidx)     * 16 + col];
      b.y = Sb[(kk + kidx + 1) * 16 + col];
      #pragma unroll
      for (int tt = 0; tt < 4; ++tt) {
        const int M0 = wave * 64 + tt * 16;
        v2f a;
        a.x = Ab[(M0 + col) * Ndim + kk + kidx];      // A-row = lane&15
        a.y = Ab[(M0 + col) * Ndim + kk + kidx + 1];
        acc[tt] = wmma_f32_16x16x4(a, b, acc[tt]);
      }
    }
    // + Bb * x_t
    #pragma unroll
    for (int tt = 0; tt < 4; ++tt) {
      const int M0 = wave * 64 + tt * 16;
      #pragma unroll
      for (int r = 0; r < 8; ++r) acc[tt][r] += Bbl[M0 + moff + r] * xv;
    }

    __syncthreads();                   // everyone done reading Sb
    #pragma unroll
    for (int tt = 0; tt < 4; ++tt) {
      const int M0 = wave * 64 + tt * 16;
      #pragma unroll
      for (int r = 0; r < 8; ++r)
        Sb[(M0 + moff + r) * 16 + col] = acc[tt][r];
    }

    if (yout) {                        // uniform branch (kernel arg)
      float pd = 0.0f;
      #pragma unroll
      for (int tt = 0; tt < 4; ++tt) {
        const int M0 = wave * 64 + tt * 16;
        #pragma unroll
        for (int r = 0; r < 8; ++r) pd += acc[tt][r] * Cl[M0 + moff + r];
      }
      pd += __shfl_xor(pd, 16, 32);    // fold the two half-waves (same col)
      if (lane < 16) atomicAdd(&yacc[col], pd);   // ds_add_f32 across 8 waves
      __syncthreads();
      if (tid < 16) {
        yout[(cgBase + tid) * CT + t] = yacc[tid];
        yacc[tid] = 0.0f;
      }
    }
    __syncthreads();                   // new state (and yacc reset) visible
  }

  if (Vout) {
    for (int e = tid; e < Ndim * 16; e += 256) {
      const int m = e >> 4, c = e & 15;
      Vout[m * NC + (cgBase + c)] = Sb[m * 16 + c];
    }
  }
}

// ---------------------------------------------------------------------------
// Kernel 4: one Hillis-Steele round over the chunk axis:
//   Hnew[:,c] = Hold[:,c] + Mk @ Hold[:,c-d]   (zero contribution if c<d)
// ---------------------------------------------------------------------------
__global__ __launch_bounds__(256) void scan_round(
    const float* __restrict__ Mk, const float* __restrict__ Hold,
    float* __restrict__ Hnew, int d, int NC)
{
  const int wave = threadIdx.x >> 5, lane = threadIdx.x & 31;
  const int col  = lane & 15;
  const int kidx = (lane >> 4) * 2;
  const int moff = (lane >> 4) * 8;
  const int M0 = blockIdx.x * 128 + wave * 16;
  const int c  = blockIdx.y * 16 + col;
  const int csrc = c - d;
  const bool vld = (csrc >= 0);
  const int cs = vld ? csrc : 0;

  prefetch_ro(&Mk[(M0 + col) * Ndim]);        // cold-start only
  prefetch_ro(&Mk[(M0 + col) * Ndim + 256]);

  v8f acc = v8f_zero();
  for (int kk = 0; kk < Ndim; kk += 4) {
    v2f a, b;
    a.x = Mk[(M0 + col) * Ndim + kk + kidx];
    a.y = Mk[(M0 + col) * Ndim + kk + kidx + 1];
    b.x = vld ? Hold[(kk + kidx)     * NC + cs] : 0.0f;
    b.y = vld ? Hold[(kk + kidx + 1) * NC + cs] : 0.0f;
    acc = wmma_f32_16x16x4(a, b, acc);
  }
  #pragma unroll
  for (int r = 0; r < 8; ++r) {
    const int m = M0 + moff + r;
    Hnew[m * NC + c] = Hold[m * NC + c] + acc[r];
  }
}

// Kernel 5: final hidden state = end state of last chunk.
__global__ void finalize_h(const float* __restrict__ H, float* __restrict__ outH, int NC) {
  const int m = blockIdx.x * blockDim.x + threadIdx.x;
  if (m < Ndim) outH[m] = H[m * NC + (NC - 1)];
}

// ---------------------------------------------------------------------------
extern "C" void kernel_launch(void* const* d_in, const int* in_sizes, int n_in,
                              void* d_out, int out_size, void* d_ws, size_t ws_size,
                              hipStream_t stream) {
  const float* x  = (const float*)d_in[0];   // (L,)
  const float* h0 = (const float*)d_in[1];   // (N,)
  const float* A  = (const float*)d_in[2];   // (N,N) lower triangular
  const float* Bv = (const float*)d_in[3];   // (N,1)
  const float* Cv = (const float*)d_in[4];   // (1,N)
  const int L  = in_sizes[0];
  const int NC = L / CT;                     // 1024 chunks for L=16384
  const float dt = 1.0f / (float)L;

  float* ws = (float*)d_ws;                  // ~7.3 MB total
  float* Ab = ws;  ws += Ndim * Ndim;
  float* Bb = ws;  ws += Ndim;
  float* HA = ws;  ws += Ndim * NC;
  float* HB = ws;  ws += Ndim * NC;
  float* P0 = ws;  ws += Ndim * Ndim;
  float* P1 = ws;  ws += Ndim * Ndim;

  // 1) discretize: Ab, Bb by triangular solve (513 columns, one wave each)
  build_ab_bb<<<(Ndim + 1 + 7) / 8, 256, 0, stream>>>(A, Bv, Ab, Bb, dt);

  // 2) pass 1: chunk-local scans -> end states V in HA
  chunk_pass<<<NC / 16, 256, 0, stream>>>(Ab, Bb, x, h0,
                                          nullptr, HA, nullptr, nullptr, NC);

  // 3) Ab^CT by repeated squaring (CT=16 -> 4 squarings), ends in P1
  dim3 sg(32, 4);
  mat_square<<<sg, 256, 0, stream>>>(Ab, P0);   // Ab^2
  mat_square<<<sg, 256, 0, stream>>>(P0, P1);   // Ab^4
  mat_square<<<sg, 256, 0, stream>>>(P1, P0);   // Ab^8
  mat_square<<<sg, 256, 0, stream>>>(P0, P1);   // Ab^16
  float* Pcur = P1; float* Palt = P0;

  // 4) Hillis-Steele inclusive scan over the chunk axis (log2(NC) rounds)
  float* Hcur = HA; float* Hnext = HB;
  for (int d = 1; d < NC; d <<= 1) {
    scan_round<<<dim3(Ndim / 128, NC / 16), 256, 0, stream>>>(Pcur, Hcur, Hnext, d, NC);
    float* ht = Hcur; Hcur = Hnext; Hnext = ht;
    if ((d << 1) < NC) {                        // Pcur <- Pcur^2 for next round
      mat_square<<<sg, 256, 0, stream>>>(Pcur, Palt);
      float* pt = Pcur; Pcur = Palt; Palt = pt;
    }
  }

  // 5) pass 2: replay chunks from scanned initial states, emit y
  float* y = (float*)d_out;
  chunk_pass<<<NC / 16, 256, 0, stream>>>(Ab, Bb, x, h0,
                                          Hcur, nullptr, Cv, y, NC);
  // 6) final hidden state
  finalize_h<<<(Ndim + 255) / 256, 256, 0, stream>>>(Hcur, y + L, NC);
}